// GAT_37873021616188
// MI455X (gfx1250) — compile-verified
//
#include <hip/hip_runtime.h>
#include <math.h>

typedef __attribute__((ext_vector_type(2))) float v2f;
typedef __attribute__((ext_vector_type(8))) float v8f;

#define F1 128   // layer-1 width (8 heads * 16 ch)
#define NH 8
#define CH 16
#define NC 40    // classes

__device__ __forceinline__ float leaky02(float v) {
    return v > 0.0f ? v : 0.2f * v;
}

// Order-preserving float atomic max via int/uint atomics.
__device__ __forceinline__ void atomicMaxF32(float* addr, float val) {
    if (val >= 0.0f)
        atomicMax((int*)addr, __float_as_int(val));
    else
        atomicMin((unsigned int*)addr, __float_as_uint(val));
}

// ---------------------------------------------------------------------------
// fill
// ---------------------------------------------------------------------------
__global__ void __launch_bounds__(256) fill_f32(float* __restrict__ p, float v, int n) {
    int i = blockIdx.x * 256 + threadIdx.x;
    if (i < n) p[i] = v;
}

// ---------------------------------------------------------------------------
// C[M,NCOLS] = A[M,128] @ B[128,NCOLS], exact fp32 via V_WMMA_F32_16X16X4_F32.
// One block per 16-row strip (M % 16 == 0). 8 waves; wave w owns column tiles
// w, w+8, ... A and B both staged in LDS with bank-conflict-free padding;
// B zero-padded to NCT16 cols so the inner loop has no guards / no addr mul.
// ---------------------------------------------------------------------------
template<int NCOLS>
__global__ void __launch_bounds__(256) gat_gemm_wmma(
    const float* __restrict__ A, const float* __restrict__ B,
    float* __restrict__ C)
{
    constexpr int K     = 128;
    constexpr int NCT16 = (NCOLS + 15) & ~15;   // padded col count
    constexpr int NCT   = NCT16 / 16;           // column tiles
    constexpr int AS    = K + 4;                // A row stride: banks 4*l16+k -> conflict-free
    constexpr int BS    = NCT16 + 8;            // B row stride: halves in disjoint bank ranges

    __shared__ float As[16 * AS];
    __shared__ float Bs[K * BS];

    const int rowBase = blockIdx.x * 16;

    // stage A: 16 x 128, contiguous in global
    for (int i = threadIdx.x; i < 16 * K; i += 256)
        As[(i >> 7) * AS + (i & 127)] = A[(size_t)rowBase * K + i];
    // stage B: 128 x NCOLS, zero-padded to NCT16
    for (int i = threadIdx.x; i < K * NCT16; i += 256) {
        int r = i / NCT16, c = i - r * NCT16;
        Bs[r * BS + c] = (c < NCOLS) ? B[r * NCOLS + c] : 0.0f;
    }
    __syncthreads();

    const int wave = threadIdx.x >> 5;
    const int lane = threadIdx.x & 31;
    const int half = lane >> 4;                 // 0: K={0,1}, 1: K={2,3}
    const int l16  = lane & 15;

    for (int ct = wave; ct < NCT; ct += 8) {
        const int colp = ct * 16 + l16;
        v8f acc = {};
        #pragma unroll 8
        for (int k = 0; k < K; k += 4) {
            // A frag (ISA 16x4 f32 layout): row = l16, K = k + 2*half + {0,1}
            v2f a = *(const v2f*)(&As[l16 * AS + k + (half << 1)]);
            v2f b;
            b.x = Bs[(k + (half << 1) + 0) * BS + colp];
            b.y = Bs[(k + (half << 1) + 1) * BS + colp];
            acc = __builtin_amdgcn_wmma_f32_16x16x4_f32(
                false, a, false, b, (short)0, acc, false, false);
        }
        if (colp < NCOLS) {
            #pragma unroll
            for (int r = 0; r < 8; ++r)
                C[(size_t)(rowBase + half * 8 + r) * NCOLS + colp] = acc[r];
        }
    }
}

// ---------------------------------------------------------------------------
// Layer-1 per-node attention logits: s_src[n,h] = <h1[n,h,:], a_src[h,:]>
// one thread per (n, head); float4 loads
// ---------------------------------------------------------------------------
__global__ void __launch_bounds__(256) att_logits_l1(
    const float* __restrict__ h, const float* __restrict__ a_src,
    const float* __restrict__ a_dst, float* __restrict__ s_src,
    float* __restrict__ s_dst, int N)
{
    int i = blockIdx.x * 256 + threadIdx.x;
    if (i >= N * NH) return;
    int n = i >> 3, head = i & 7;
    const float4* hp = (const float4*)(h + (size_t)n * F1 + head * CH);
    const float4* as = (const float4*)(a_src + head * CH);
    const float4* ad = (const float4*)(a_dst + head * CH);
    float ss = 0.f, sd = 0.f;
    #pragma unroll
    for (int q = 0; q < CH / 4; ++q) {
        float4 v = hp[q], a = as[q], d = ad[q];
        ss += v.x * a.x + v.y * a.y + v.z * a.z + v.w * a.w;
        sd += v.x * d.x + v.y * d.y + v.z * d.z + v.w * d.w;
    }
    s_src[i] = ss;
    s_dst[i] = sd;
}

// ---------------------------------------------------------------------------
// Layer-2 per-node attention logits (1 head, 40 ch): one thread per node
// ---------------------------------------------------------------------------
__global__ void __launch_bounds__(256) att_logits_l2(
    const float* __restrict__ g, const float* __restrict__ a_src,
    const float* __restrict__ a_dst, float* __restrict__ s_src,
    float* __restrict__ s_dst, int N)
{
    int n = blockIdx.x * 256 + threadIdx.x;
    if (n >= N) return;
    float ss = 0.f, sd = 0.f;
    #pragma unroll
    for (int c = 0; c < NC; ++c) {
        float v = g[(size_t)n * NC + c];
        ss += v * a_src[c];
        sd += v * a_dst[c];
    }
    s_src[n] = ss;
    s_dst[n] = sd;
}

// ---------------------------------------------------------------------------
// Layer-1 segment max over destinations: one thread per (edge, head).
// Edges E..E+N-1 are self-loops.
// ---------------------------------------------------------------------------
__global__ void __launch_bounds__(256) edge_max_l1(
    const int* __restrict__ src, const int* __restrict__ dst, int E, int N,
    const float* __restrict__ s_src, const float* __restrict__ s_dst,
    float* __restrict__ m)
{
    int i = blockIdx.x * 256 + threadIdx.x;
    if (i >= (E + N) * NH) return;
    int e = i >> 3, head = i & 7;
    int s, d;
    if (e < E) { s = src[e]; d = dst[e]; } else { s = d = e - E; }
    float v = leaky02(s_src[s * NH + head] + s_dst[d * NH + head]);
    atomicMaxF32(&m[d * NH + head], v);
}

// ---------------------------------------------------------------------------
// Layer-1 weighted scatter: ONE WAVE PER EDGE.
// Lane l owns channels 4l..4l+3 (head = l/4): single b128 gather of h[src],
// softmax weight p computed once per lane, 4 L2 atomics.
// Normalization deferred (den constant per (dst,head)).
// ---------------------------------------------------------------------------
__global__ void __launch_bounds__(256) edge_acc_l1(
    const int* __restrict__ src, const int* __restrict__ dst, int E, int N,
    const float* __restrict__ s_src, const float* __restrict__ s_dst,
    const float* __restrict__ m, const float* __restrict__ h,
    float* __restrict__ acc, float* __restrict__ den)
{
    int wid = (blockIdx.x * 256 + threadIdx.x) >> 5;   // edge id (wave-uniform)
    if (wid >= E + N) return;
    const int lane = threadIdx.x & 31;
    int s, d;
    if (wid < E) { s = src[wid]; d = dst[wid]; } else { s = d = wid - E; }
    const int head = lane >> 2;
    float logit = leaky02(s_src[s * NH + head] + s_dst[d * NH + head]);
    float p = __expf(logit - m[d * NH + head]);
    const int c0 = lane << 2;
    float4 hv = *(const float4*)(h + (size_t)s * F1 + c0);
    float* ap = acc + (size_t)d * F1 + c0;
    atomicAdd(ap + 0, p * hv.x);
    atomicAdd(ap + 1, p * hv.y);
    atomicAdd(ap + 2, p * hv.z);
    atomicAdd(ap + 3, p * hv.w);
    if ((lane & 3) == 0) atomicAdd(&den[d * NH + head], p);
}

// ---------------------------------------------------------------------------
// Layer-1 finalize: acc = relu(acc/den + b1)  (in place; becomes L2 input)
// ---------------------------------------------------------------------------
__global__ void __launch_bounds__(256) finalize_l1(
    float* __restrict__ acc, const float* __restrict__ den,
    const float* __restrict__ b1, int N)
{
    int i = blockIdx.x * 256 + threadIdx.x;
    if (i >= N * F1) return;
    int n = i >> 7, c = i & 127;
    float v = acc[i] / (den[n * NH + (c >> 4)] + 1e-16f) + b1[c];
    acc[i] = v > 0.f ? v : 0.f;
}

// ---------------------------------------------------------------------------
// Layer-2 segment max: one thread per edge (1 head)
// ---------------------------------------------------------------------------
__global__ void __launch_bounds__(256) edge_max_l2(
    const int* __restrict__ src, const int* __restrict__ dst, int E, int N,
    const float* __restrict__ s_src, const float* __restrict__ s_dst,
    float* __restrict__ m)
{
    int e = blockIdx.x * 256 + threadIdx.x;
    if (e >= E + N) return;
    int s, d;
    if (e < E) { s = src[e]; d = dst[e]; } else { s = d = e - E; }
    float v = leaky02(s_src[s] + s_dst[d]);
    atomicMaxF32(&m[d], v);
}

// ---------------------------------------------------------------------------
// Layer-2 weighted scatter: 8 threads per edge, 5 classes each.
// ---------------------------------------------------------------------------
__global__ void __launch_bounds__(256) edge_acc_l2(
    const int* __restrict__ src, const int* __restrict__ dst, int E, int N,
    const float* __restrict__ s_src, const float* __restrict__ s_dst,
    const float* __restrict__ m, const float* __restrict__ g,
    float* __restrict__ acc, float* __restrict__ den)
{
    int t = blockIdx.x * 256 + threadIdx.x;
    int e = t >> 3, sub = t & 7;
    if (e >= E + N) return;
    int s, d;
    if (e < E) { s = src[e]; d = dst[e]; } else { s = d = e - E; }
    float p = __expf(leaky02(s_src[s] + s_dst[d]) - m[d]);
    const int c0 = sub * 5;
    const float* gp = g + (size_t)s * NC + c0;
    float* ap = acc + (size_t)d * NC + c0;
    #pragma unroll
    for (int j = 0; j < 5; ++j)
        atomicAdd(ap + j, p * gp[j]);
    if (sub == 0) atomicAdd(&den[d], p);
}

// ---------------------------------------------------------------------------
// Layer-2 finalize + log_softmax: one thread per node (40 regs of z)
// ---------------------------------------------------------------------------
__global__ void __launch_bounds__(256) final_lsm(
    const float* __restrict__ acc, const float* __restrict__ den,
    const float* __restrict__ b2, float* __restrict__ out, int N)
{
    int n = blockIdx.x * 256 + threadIdx.x;
    if (n >= N) return;
    float dinv = 1.0f / (den[n] + 1e-16f);
    float z[NC];
    float mx = -__builtin_inff();
    #pragma unroll
    for (int c = 0; c < NC; ++c) {
        z[c] = acc[(size_t)n * NC + c] * dinv + b2[c];
        mx = fmaxf(mx, z[c]);
    }
    float s = 0.f;
    #pragma unroll
    for (int c = 0; c < NC; ++c) s += __expf(z[c] - mx);
    float lse = mx + __logf(s);
    #pragma unroll
    for (int c = 0; c < NC; ++c)
        out[(size_t)n * NC + c] = z[c] - lse;
}

// ---------------------------------------------------------------------------
extern "C" void kernel_launch(void* const* d_in, const int* in_sizes, int n_in,
                              void* d_out, int out_size, void* d_ws, size_t ws_size,
                              hipStream_t stream) {
    const float* x      = (const float*)d_in[0];
    const int*   edge   = (const int*)  d_in[1];
    const float* W1     = (const float*)d_in[2];
    const float* asrc1  = (const float*)d_in[3];
    const float* adst1  = (const float*)d_in[4];
    const float* b1     = (const float*)d_in[5];
    const float* W2     = (const float*)d_in[6];
    const float* asrc2  = (const float*)d_in[7];
    const float* adst2  = (const float*)d_in[8];
    const float* b2     = (const float*)d_in[9];
    float* out = (float*)d_out;

    const int N = in_sizes[0] / F1;     // 100000
    const int E = in_sizes[1] / 2;      // 1600000
    const int* src = edge;
    const int* dst = edge + E;

    // workspace carve-up (~149 MB of fp32)
    float* ws   = (float*)d_ws;
    float* h1   = ws; ws += (size_t)N * F1;   // x @ W1
    float* ss1  = ws; ws += (size_t)N * NH;
    float* sd1  = ws; ws += (size_t)N * NH;
    float* m1   = ws; ws += (size_t)N * NH;
    float* den1 = ws; ws += (size_t)N * NH;
    float* acc1 = ws; ws += (size_t)N * F1;   // L1 agg -> relu -> L2 input
    float* g2   = ws; ws += (size_t)N * NC;   // acc1 @ W2
    float* ss2  = ws; ws += N;
    float* sd2  = ws; ws += N;
    float* m2   = ws; ws += N;
    float* den2 = ws; ws += N;
    float* acc2 = ws; ws += (size_t)N * NC;

    const float NINF = -__builtin_inff();
    auto blocks = [](long long n) { return (unsigned)((n + 255) / 256); };

    // init accumulators every launch (deterministic replay)
    fill_f32<<<blocks((long long)N * NH), 256, 0, stream>>>(m1,   NINF, N * NH);
    fill_f32<<<blocks((long long)N * NH), 256, 0, stream>>>(den1, 0.f,  N * NH);
    fill_f32<<<blocks((long long)N * F1), 256, 0, stream>>>(acc1, 0.f,  N * F1);
    fill_f32<<<blocks(N),                 256, 0, stream>>>(m2,   NINF, N);
    fill_f32<<<blocks(N),                 256, 0, stream>>>(den2, 0.f,  N);
    fill_f32<<<blocks((long long)N * NC), 256, 0, stream>>>(acc2, 0.f,  N * NC);

    // ---- layer 1 ----
    gat_gemm_wmma<F1><<<N / 16, 256, 0, stream>>>(x, W1, h1);
    att_logits_l1<<<blocks((long long)N * NH), 256, 0, stream>>>(h1, asrc1, adst1, ss1, sd1, N);
    edge_max_l1<<<blocks((long long)(E + N) * NH), 256, 0, stream>>>(src, dst, E, N, ss1, sd1, m1);
    edge_acc_l1<<<blocks((long long)(E + N) * 32), 256, 0, stream>>>(src, dst, E, N, ss1, sd1, m1, h1, acc1, den1);
    finalize_l1<<<blocks((long long)N * F1), 256, 0, stream>>>(acc1, den1, b1, N);

    // ---- layer 2 ----
    gat_gemm_wmma<NC><<<N / 16, 256, 0, stream>>>(acc1, W2, g2);
    att_logits_l2<<<blocks(N), 256, 0, stream>>>(g2, asrc2, adst2, ss2, sd2, N);
    edge_max_l2<<<blocks(E + N), 256, 0, stream>>>(src, dst, E, N, ss2, sd2, m2);
    edge_acc_l2<<<blocks((long long)(E + N) * 8), 256, 0, stream>>>(src, dst, E, N, ss2, sd2, m2, g2, acc2, den2);
    final_lsm<<<blocks(N), 256, 0, stream>>>(acc2, den2, b2, out, N);
}